// LazyGCNConv_77025943487121
// MI455X (gfx1250) — compile-verified
//
#include <hip/hip_runtime.h>
#include <hip/hip_bf16.h>

typedef __attribute__((ext_vector_type(2))) float v2f;
typedef __attribute__((ext_vector_type(8))) float v8f;

#define D_FEAT 128
#define A_PITCH 132   // 128 + 4 pad: kills LDS bank conflicts on stride-128 row reads

// ---------------------------------------------------------------------------
// K1: zero the integer degree array
// ---------------------------------------------------------------------------
__global__ void k_zero_i32(int* __restrict__ p, int n) {
    int i = blockIdx.x * blockDim.x + threadIdx.x;
    if (i < n) p[i] = 0;
}

// ---------------------------------------------------------------------------
// K2: degree count on destination nodes; original self-loops get weight 0
// ---------------------------------------------------------------------------
__global__ void k_count_deg(const int* __restrict__ ei, int E, int* __restrict__ deg) {
    int e = blockIdx.x * blockDim.x + threadIdx.x;
    if (e >= E) return;
    int r = ei[e];
    int c = ei[E + e];
    if (r != c) atomicAdd(&deg[c], 1);
}

// ---------------------------------------------------------------------------
// K3: dis[n] = rsqrt(deg[n] + 1)   (the +1 is the appended self-loop; deg>0 always)
// ---------------------------------------------------------------------------
__global__ void k_dis(const int* __restrict__ deg, float* __restrict__ dis, int N) {
    int i = blockIdx.x * blockDim.x + threadIdx.x;
    if (i < N) dis[i] = rsqrtf(1.0f + (float)deg[i]);
}

// ---------------------------------------------------------------------------
// K4: xw = x @ W via V_WMMA_F32_16X16X4_F32.
// Block = 256 threads = 8 waves. Block computes rows [row0, row0+16) x all 128 cols;
// wave w owns the 16-column tile starting at n0 = 16*w.
// A fragment (ISA 7.12.2, 32-bit A 16x4): lane L holds M = L%16,
//   vector elem j holds K = k0 + j + 2*(L/16)  -> contiguous float2 per lane.
// B fragment (4x16, row-striped across lanes): lane L holds N = L%16,
//   elem j holds K = k0 + j + 2*(L/16).
// C/D fragment: elem v -> row M = v + 8*(L/16), col N = L%16.
// ---------------------------------------------------------------------------
__global__ __launch_bounds__(256)
void k_gemm_xw_wmma(const float* __restrict__ x, const float* __restrict__ w,
                    float* __restrict__ xw, int n_rows) {
    __shared__ float sA[16 * A_PITCH];   // 16 rows of x, padded pitch

    const int tid  = threadIdx.x;
    const int wave = tid >> 5;           // 0..7  -> column tile
    const int lane = tid & 31;
    const int row0 = blockIdx.x * 16;

    // Cooperative float4 load of the 16x128 A slab (512 float4 / 256 threads).
    for (int i = tid; i < 16 * 32; i += 256) {
        const int r  = i >> 5;           // row within slab
        const int c4 = i & 31;           // float4 column
        float4 v = make_float4(0.f, 0.f, 0.f, 0.f);
        if (row0 + r < n_rows)
            v = ((const float4*)(x + (size_t)(row0 + r) * D_FEAT))[c4];
        float* dst = &sA[r * A_PITCH + c4 * 4];
        dst[0] = v.x; dst[1] = v.y; dst[2] = v.z; dst[3] = v.w;
    }
    __syncthreads();

    const int hi = lane >> 4;            // 0 or 1 (half-wave)
    const int lo = lane & 15;
    const int n0 = wave * 16;

    v8f c = {};
    #pragma unroll
    for (int k0 = 0; k0 < D_FEAT; k0 += 4) {
        const int k = k0 + 2 * hi;
        // A fragment: ds_load_b64, conflict-free thanks to A_PITCH=132
        v2f a = *(const v2f*)&sA[lo * A_PITCH + k];
        // B fragment: two coalesced 64B row reads of W (resident in L2/L0)
        v2f b;
        b.x = w[(size_t)k * D_FEAT + n0 + lo];
        b.y = w[(size_t)(k + 1) * D_FEAT + n0 + lo];
        c = __builtin_amdgcn_wmma_f32_16x16x4_f32(
                /*neg_a=*/false, a, /*neg_b=*/false, b,
                /*c_mod=*/(short)0, c, /*reuse_a=*/false, /*reuse_b=*/false);
    }

    // Store C fragment: lanes 0-15 write contiguous 64B per row -> coalesced.
    float* outp = xw + (size_t)row0 * D_FEAT + n0 + lo;
    #pragma unroll
    for (int v = 0; v < 8; ++v) {
        const int m = v + 8 * hi;
        if (row0 + m < n_rows)
            outp[(size_t)m * D_FEAT] = c[v];
    }
}

// ---------------------------------------------------------------------------
// K5: out[n,:] = bias + dis[n]^2 * xw[n,:]   (self-loop contribution, float4)
// Thread i handles node i>>5, float4 chunk i&31.
// ---------------------------------------------------------------------------
__global__ void k_init_out(const float* __restrict__ xw, const float* __restrict__ dis,
                           const float* __restrict__ bias, float* __restrict__ out, int N) {
    const long long i = (long long)blockIdx.x * blockDim.x + threadIdx.x;
    if (i >= (long long)N * 32) return;
    const int n  = (int)(i >> 5);
    const int c4 = (int)(i & 31);
    float s = dis[n];
    s = s * s;
    const float4 v = ((const float4*)(xw + (size_t)n * D_FEAT))[c4];
    const float4 b = ((const float4*)bias)[c4];
    float4 o;
    o.x = b.x + s * v.x;
    o.y = b.y + s * v.y;
    o.z = b.z + s * v.z;
    o.w = b.w + s * v.w;
    ((float4*)(out + (size_t)n * D_FEAT))[c4] = o;
}

// ---------------------------------------------------------------------------
// K6: edge scatter. One wave (32 lanes) per edge; lane handles one float4:
// coalesced 512B gather of xw[row], 4x global_atomic_add_f32 into out[col].
// ---------------------------------------------------------------------------
__global__ void k_scatter(const int* __restrict__ ei, int E,
                          const float* __restrict__ xw, const float* __restrict__ dis,
                          float* __restrict__ out) {
    const long long idx = (long long)blockIdx.x * blockDim.x + threadIdx.x;
    const int e    = (int)(idx >> 5);
    const int lane = (int)(idx & 31);
    if (e >= E) return;
    const int r = ei[e];
    const int c = ei[E + e];
    if (r == c) return;                       // original self-loop: weight 0
    const float norm = dis[r] * dis[c];
    const float4 v = ((const float4*)(xw + (size_t)r * D_FEAT))[lane];
    float* dst = out + (size_t)c * D_FEAT + lane * 4;
    atomicAdd(dst + 0, norm * v.x);
    atomicAdd(dst + 1, norm * v.y);
    atomicAdd(dst + 2, norm * v.z);
    atomicAdd(dst + 3, norm * v.w);
}

// ---------------------------------------------------------------------------
extern "C" void kernel_launch(void* const* d_in, const int* in_sizes, int n_in,
                              void* d_out, int out_size, void* d_ws, size_t ws_size,
                              hipStream_t stream) {
    const float* x    = (const float*)d_in[0];
    const int*   ei   = (const int*)  d_in[1];
    const float* w    = (const float*)d_in[2];
    const float* bias = (const float*)d_in[3];
    float*       out  = (float*)d_out;

    const int N = in_sizes[0] / D_FEAT;   // 100000
    const int E = in_sizes[1] / 2;        // 1600000

    // Workspace layout: [deg int32 x N][dis f32 x N][xw f32 x N*128]
    char*  ws   = (char*)d_ws;
    int*   deg  = (int*)ws;
    float* dis  = (float*)(ws + (size_t)N * 4);
    size_t xwof = (((size_t)N * 8) + 255) & ~(size_t)255;
    float* xw   = (float*)(ws + xwof);

    k_zero_i32 <<<(N + 255) / 256, 256, 0, stream>>>(deg, N);
    k_count_deg<<<(E + 255) / 256, 256, 0, stream>>>(ei, E, deg);
    k_dis      <<<(N + 255) / 256, 256, 0, stream>>>(deg, dis, N);
    k_gemm_xw_wmma<<<(N + 15) / 16, 256, 0, stream>>>(x, w, xw, N);

    const long long initThreads = (long long)N * 32;
    k_init_out <<<(unsigned)((initThreads + 255) / 256), 256, 0, stream>>>(xw, dis, bias, out, N);

    const long long scatThreads = (long long)E * 32;
    k_scatter  <<<(unsigned)((scatThreads + 255) / 256), 256, 0, stream>>>(ei, E, xw, dis, out);
}